// Binary_Encoder_11708080849219
// MI455X (gfx1250) — compile-verified
//
#include <hip/hip_runtime.h>
#include <stdint.h>

typedef float    f32x4 __attribute__((ext_vector_type(4)));
typedef unsigned u32x4 __attribute__((ext_vector_type(4)));
typedef int      i32x4 __attribute__((ext_vector_type(4)));
typedef int      i32x8 __attribute__((ext_vector_type(8)));

#define ROWS_PER_BLOCK 8
#define CH 512
#define NBITS 10

#if defined(__HIP_DEVICE_COMPILE__) && __has_builtin(__builtin_amdgcn_tensor_load_to_lds)
#define USE_TDM 1
#else
#define USE_TDM 0
#endif

// Binary SAR encoder: out[row, bit, ch] = bit (9-bit) of floor(clamp(x,0,1)*1024) clamped to 1023.
// One wave per 512-channel row; input staged via TDM (tensor_load_to_lds) into LDS,
// output streamed with 128-bit non-temporal stores.
__global__ __launch_bounds__(256) void Binary_Encoder_kernel(const float* __restrict__ x,
                                                             unsigned* __restrict__ out,
                                                             int rows) {
    __shared__ __align__(16) float tile[ROWS_PER_BLOCK * CH];
    const int wave = threadIdx.x >> 5;
    const int lane = threadIdx.x & 31;
    const int row  = blockIdx.x * ROWS_PER_BLOCK + wave;
    if (row >= rows) return;

#if USE_TDM
    {
        // ---- Build Tensor DMA Descriptor (D#), wave-uniform in SGPRs ----
        const uint64_t ga   = (uint64_t)(uintptr_t)x + (uint64_t)row * (CH * 4u);
        // Low 32 bits of a generic (flat) LDS pointer == byte offset in the WG's LDS allocation.
        const unsigned ldsa = (unsigned)(uintptr_t)(&tile[wave * CH]);

        u32x4 g0;
        g0.x = __builtin_amdgcn_readfirstlane(1u);                 // count=1, user mode, no gather
        g0.y = __builtin_amdgcn_readfirstlane(ldsa);               // lds_addr [63:32]
        g0.z = __builtin_amdgcn_readfirstlane((unsigned)ga);       // global_addr [95:64]
        g0.w = __builtin_amdgcn_readfirstlane(
                   (((unsigned)(ga >> 32)) & 0x01FFFFFFu) | (2u << 30)); // addr[56:32] | type=2

        const unsigned urows = (unsigned)rows;
        i32x8 g1;
        g1[0] = __builtin_amdgcn_readfirstlane((int)0x00020000u);          // data_size=2 (4B), no flags
        g1[1] = __builtin_amdgcn_readfirstlane((int)((unsigned)CH << 16)); // tensor_dim0=512 @bit48
        g1[2] = __builtin_amdgcn_readfirstlane((int)((urows & 0xFFFFu) << 16));          // tensor_dim1.lo
        g1[3] = __builtin_amdgcn_readfirstlane((int)(((urows >> 16) & 0xFFFFu)
                                                    | ((unsigned)CH << 16)));            // tensor_dim1.hi | tile_dim0=512
        g1[4] = __builtin_amdgcn_readfirstlane(1);                         // tile_dim1=1, tile_dim2=0
        g1[5] = __builtin_amdgcn_readfirstlane(CH);                        // tensor_dim0_stride=512
        g1[6] = __builtin_amdgcn_readfirstlane(0);
        g1[7] = __builtin_amdgcn_readfirstlane(0);

        const i32x4 z4 = {0, 0, 0, 0};
#if defined(__clang_major__) && __clang_major__ >= 23
        const i32x8 z8 = {0, 0, 0, 0, 0, 0, 0, 0};
        __builtin_amdgcn_tensor_load_to_lds(g0, g1, z4, z4, z8, 0);
#else
        __builtin_amdgcn_tensor_load_to_lds(g0, g1, z4, z4, 0);
#endif
        __builtin_amdgcn_s_wait_tensorcnt(0);
    }
    const f32x4* __restrict__ src = (const f32x4*)(&tile[wave * CH]);
#else
    const f32x4* __restrict__ src = (const f32x4*)(x + (size_t)row * CH);
#endif

    unsigned* __restrict__ orow = out + (size_t)row * (NBITS * CH);

#pragma unroll
    for (int it = 0; it < 4; ++it) {
        const int vec = it * 32 + lane;        // float4 index within the row (0..127)
        f32x4 v = src[vec];

        unsigned q[4];
#pragma unroll
        for (int j = 0; j < 4; ++j) {
            float c = fminf(fmaxf(v[j], 0.0f), 1.0f);
            unsigned t = (unsigned)(c * 1024.0f);   // exact: power-of-two scale + truncate
            q[j] = t > 1023u ? 1023u : t;           // v==1.0 -> all ten bits set
        }

        unsigned* obase = orow + (size_t)vec * 4;   // channel offset = vec*4
#pragma unroll
        for (int i = 0; i < NBITS; ++i) {
            const int sh = 9 - i;
            u32x4 b;
            b.x = ((q[0] >> sh) & 1u) ? 0x3F800000u : 0u;
            b.y = ((q[1] >> sh) & 1u) ? 0x3F800000u : 0u;
            b.z = ((q[2] >> sh) & 1u) ? 0x3F800000u : 0u;
            b.w = ((q[3] >> sh) & 1u) ? 0x3F800000u : 0u;
            __builtin_nontemporal_store(b, (u32x4*)(obase + (size_t)i * CH));
        }
    }
}

extern "C" void kernel_launch(void* const* d_in, const int* in_sizes, int n_in,
                              void* d_out, int out_size, void* d_ws, size_t ws_size,
                              hipStream_t stream) {
    const float* x  = (const float*)d_in[0];
    unsigned* out   = (unsigned*)d_out;        // fp32 bits written as raw 0x0 / 0x3F800000
    const int n     = in_sizes[0];             // 16*1024*512
    const int rows  = n / CH;                  // 16384
    const int grid  = (rows + ROWS_PER_BLOCK - 1) / ROWS_PER_BLOCK;
    Binary_Encoder_kernel<<<dim3(grid), dim3(256), 0, stream>>>(x, out, rows);
}